// GPGmodel_without_NN_38208029065944
// MI455X (gfx1250) — compile-verified
//
#include <hip/hip_runtime.h>
#include <hip/hip_bf16.h>

typedef __attribute__((ext_vector_type(2))) float v2f;
typedef __attribute__((ext_vector_type(8))) float v8f;

#define N_BUS 118

// ---------------------------------------------------------------------------
// p[i] = x[i,0] - x[i,1];  denom[i] = ybus[b, r, r] * 100  (b = i/118, r = i%118)
// ---------------------------------------------------------------------------
__global__ void init_kernel(const float* __restrict__ x,
                            const float* __restrict__ ybus,
                            float* __restrict__ p,
                            float* __restrict__ denom,
                            int n) {
    int i = blockIdx.x * blockDim.x + threadIdx.x;
    if (i < n) {
        p[i] = x[2 * i] - x[2 * i + 1];
        int b = i / N_BUS;
        int r = i - b * N_BUS;
        denom[i] = ybus[(size_t)b * N_BUS * N_BUS + (size_t)r * N_BUS + r] * 100.0f;
    }
}

__global__ void zero_kernel(float* __restrict__ buf, int n) {
    int i = blockIdx.x * blockDim.x + threadIdx.x;
    if (i < n) buf[i] = 0.0f;
}

// ---------------------------------------------------------------------------
// Scatter: aggr[dst[e]] += theta[src[e]] * (w[e] * 100)
// edge_index is [2, nE] row-major int64: src at [0..nE), dst at [nE..2nE).
// One thread per edge; index/weight streams are unit-stride (coalesced) and
// L2-resident after the first layer; aggr (1.9 MB) lives in L2.
// ---------------------------------------------------------------------------
__global__ void scatter_kernel(const long long* __restrict__ ei,
                               const float* __restrict__ ew,
                               const float* __restrict__ theta,
                               float* __restrict__ aggr,
                               int nE) {
    int e = blockIdx.x * blockDim.x + threadIdx.x;
    if (e < nE) {
        int src = (int)ei[e];
        int dst = (int)ei[(size_t)nE + e];
        float msg = theta[src] * (ew[e] * 100.0f);
        atomicAdd(&aggr[dst], msg);
    }
}

// ---------------------------------------------------------------------------
// out[b*118 + t] = (p - aggr)/denom  minus the slack-bus (t==0) value.
// One block (128 threads) per graph (118 buses) so the slack value is shared via LDS.
// ---------------------------------------------------------------------------
__global__ void gpg_out_kernel(const float* __restrict__ p,
                               const float* __restrict__ aggr,
                               const float* __restrict__ denom,
                               float* __restrict__ out) {
    __shared__ float ref;
    int t = threadIdx.x;
    int i = blockIdx.x * N_BUS + t;
    float v = 0.0f;
    if (t < N_BUS) v = (p[i] - aggr[i]) / denom[i];
    if (t == 0) ref = v;
    __syncthreads();
    if (t < N_BUS) out[i] = v - ref;
}

// ---------------------------------------------------------------------------
// err += sum_i |p[i] - aggr[i]|   (exact f32 via V_WMMA_F32_16X16X4_F32)
//
// A = ones(16x4), so D = A*B + C accumulates per-column sums of each 4x16
// f32 tile; the mapping of the 64 wave values into tile slots is irrelevant
// for the grand total. Main loop handles only full 64-element chunks (no
// in-loop predication; N is a multiple of 64 here), each lane contributing
// a single float2 (b64) load per array. c[0] lanes 16-31 duplicate lanes
// 0-15 (row M=8 == row M=0), so mask them off, fold in the generic tail,
// and do one full 32-lane xor-tree + one atomicAdd per wave.
// ---------------------------------------------------------------------------
__global__ void error_wmma_kernel(const float* __restrict__ p,
                                  const float* __restrict__ aggr,
                                  int n,
                                  float* __restrict__ err) {
    int tid  = blockIdx.x * blockDim.x + threadIdx.x;
    int lane = threadIdx.x & 31;
    int wave = tid >> 5;
    int nwaves = (gridDim.x * blockDim.x) >> 5;
    int step = nwaves * 64;

    v2f a;
    a.x = 1.0f;
    a.y = 1.0f;
    v8f c = {};

    for (int base = wave * 64; base + 64 <= n; base += step) {
        // next chunk prefetch (near-cache hint; speculative, safely dropped)
        __builtin_prefetch(p + base + step + 2 * lane, 0, 3);
        __builtin_prefetch(aggr + base + step + 2 * lane, 0, 3);

        v2f pv = *(const v2f*)(p + base + 2 * lane);
        v2f av = *(const v2f*)(aggr + base + 2 * lane);
        v2f b;
        b.x = fabsf(pv.x - av.x);
        b.y = fabsf(pv.y - av.y);
        // 8 args: (neg_a, A, neg_b, B, c_mod, C, reuse_a, reuse_b)
        c = __builtin_amdgcn_wmma_f32_16x16x4_f32(
                false, a, false, b, (short)0, c, false, false);
    }

    // lanes 0-15 hold the 16 distinct column sums; 16-31 are duplicates
    float v = (lane < 16) ? c[0] : 0.0f;

    // generic tail (n % 64 elements), handled by the grid's first threads
    int ntail = n & ~63;
    int rem = n - ntail;
    if (tid < rem) v += fabsf(p[ntail + tid] - aggr[ntail + tid]);

    #pragma unroll
    for (int off = 1; off < 32; off <<= 1)
        v += __shfl_xor(v, off, 32);
    if (lane == 0) atomicAdd(err, v);
}

// ---------------------------------------------------------------------------
// Orchestration: 11 gpg_layer evaluations + 11 lc_error evaluations.
// Layer 0 has theta == 0, so its scatter contributes nothing (skip it).
// ---------------------------------------------------------------------------
extern "C" void kernel_launch(void* const* d_in, const int* in_sizes, int n_in,
                              void* d_out, int out_size, void* d_ws, size_t ws_size,
                              hipStream_t stream) {
    const float*     x     = (const float*)d_in[0];
    // d_in[1] = y (only used for shape)
    const long long* ei_nd = (const long long*)d_in[2];
    const float*     ea_nd = (const float*)d_in[3];
    const long long* ei    = (const long long*)d_in[4];
    const float*     ea    = (const float*)d_in[5];
    const float*     ybus  = (const float*)d_in[6];

    const int N    = in_sizes[1];        // 483328
    const int E_ND = in_sizes[3];        // 1933312
    const int E    = in_sizes[5];        // 2416640
    const int B    = N / N_BUS;          // 4096

    float* out = (float*)d_out;          // [N] theta, then [11] errors
    float* err = out + N;

    float* ws    = (float*)d_ws;
    float* p     = ws;                   // [N]
    float* denom = ws + (size_t)N;       // [N]
    float* aggr  = ws + (size_t)2 * N;   // [N]

    const int T = 256;
    dim3 gN((N + T - 1) / T);
    dim3 gEnd((E_ND + T - 1) / T);
    dim3 gE((E + T - 1) / T);
    const int RED_BLOCKS = 512;          // 4096 waves for the reduction

    init_kernel<<<gN, T, 0, stream>>>(x, ybus, p, denom, N);
    zero_kernel<<<1, 32, 0, stream>>>(err, 11);

    // ---- layer 0: theta = 0  =>  aggr = 0, out = p/denom - slack ----
    zero_kernel<<<gN, T, 0, stream>>>(aggr, N);
    gpg_out_kernel<<<B, 128, 0, stream>>>(p, aggr, denom, out);

    // ---- error 0 ----
    zero_kernel<<<gN, T, 0, stream>>>(aggr, N);
    scatter_kernel<<<gE, T, 0, stream>>>(ei, ea, out, aggr, E);
    error_wmma_kernel<<<RED_BLOCKS, T, 0, stream>>>(p, aggr, N, err + 0);

    // ---- layers 1..10 ----
    for (int k = 1; k <= 10; ++k) {
        zero_kernel<<<gN, T, 0, stream>>>(aggr, N);
        scatter_kernel<<<gEnd, T, 0, stream>>>(ei_nd, ea_nd, out, aggr, E_ND);
        gpg_out_kernel<<<B, 128, 0, stream>>>(p, aggr, denom, out);

        zero_kernel<<<gN, T, 0, stream>>>(aggr, N);
        scatter_kernel<<<gE, T, 0, stream>>>(ei, ea, out, aggr, E);
        error_wmma_kernel<<<RED_BLOCKS, T, 0, stream>>>(p, aggr, N, err + k);
    }
}